// PCEN_53730040873504
// MI455X (gfx1250) — compile-verified
//
#include <hip/hip_runtime.h>
#include <hip/hip_bf16.h>

typedef __attribute__((ext_vector_type(2))) float v2f;
typedef __attribute__((ext_vector_type(4))) float v4f;
typedef __attribute__((ext_vector_type(8))) float v8f;

#define PCEN_EPS 1e-6f
#define CCH 128
#define TLEN 8192
#define NROWS 4096            // B*C = 32*128
#define WAVES_PER_BLOCK 8

__device__ __forceinline__ float fexp2(float x) {
#if __has_builtin(__builtin_amdgcn_exp2f)
    return __builtin_amdgcn_exp2f(x);      // v_exp_f32 (2^x)
#else
    return exp2f(x);
#endif
}

__device__ __forceinline__ float flog2(float x) {
#if __has_builtin(__builtin_amdgcn_logf)
    return __builtin_amdgcn_logf(x);       // v_log_f32 (log2)
#else
    return log2f(x);
#endif
}

// One wave32 per (b,c) row. Per 256-element tile:
//   P(16x16) = M(16x16) * X(16x16) via 4x V_WMMA_F32_16X16X4_F32 (K accumulation)
// where M[i][j] = w*(1-w)^(i-j) (lower triangular, per-channel constant),
// X[j][n] = x[tile_base + 16n + j]  (chunk n, position j).
// P[i][n] is the zero-carry chunk EMA; true EMA: e = P[i][n] + (1-w)^(i+1)*carry_n.
// Chunk carries: affine scan c_{n+1} = d16*c_n + P[15][n] done with lane shuffles.
__global__ __launch_bounds__(256) void pcen_wmma_kernel(
    const float* __restrict__ x,
    const float* __restrict__ alpha,
    const float* __restrict__ delta,
    const float* __restrict__ root,
    const float* __restrict__ emaw,
    float* __restrict__ out)
{
    const int lane = threadIdx.x & 31;
    const int wave = threadIdx.x >> 5;
    const int row  = blockIdx.x * WAVES_PER_BLOCK + wave;   // 0..4095
    const int c    = row & (CCH - 1);

    const int half = lane >> 4;      // which 16-lane half
    const int nn   = lane & 15;      // chunk index n (for B/D), row m (for A)

    // per-channel parameters (uniform across the wave)
    const float w     = fminf(fmaxf(emaw[c], 0.0f), 0.2f);
    const float a     = fminf(alpha[c], 1.0f);
    const float r     = fmaxf(root[c], 1.0f);
    const float inv_r = 1.0f / r;
    const float del   = delta[c];
    const float dpr   = fexp2(inv_r * flog2(del));   // delta^(1/r)
    const float d     = 1.0f - w;

    float d16;
    {
        float t2 = d * d, t4 = t2 * t2, t8 = t4 * t4;
        d16 = t8 * t8;
    }

    // dp[v] = d^(m+1) for m = v + 8*half  ->  d^(1+8*half) ... d^(8+8*half)
    float dp[8];
    {
        float b0 = d;
        if (half) { float t2 = d*d, t4 = t2*t2, t8 = t4*t4; b0 = d * t8; } // d^9
        dp[0] = b0;
#pragma unroll
        for (int i = 1; i < 8; ++i) dp[i] = dp[i - 1] * d;
    }

    // A fragments (constant per row): lane holds M[m][k], m = nn, k = 4g + 2*half + {0,1}
    const float lgd = flog2(d);
    v2f afrag[4];
#pragma unroll
    for (int g = 0; g < 4; ++g) {
        const int k0 = 4 * g + 2 * half;
        float e0 = 0.0f, e1 = 0.0f;
        if (k0     <= nn) e0 = w * fexp2((float)(nn - k0) * lgd);     // w*d^(nn-k0)
        if (k0 + 1 <= nn) e1 = w * fexp2((float)(nn - k0 - 1) * lgd);
        afrag[g].x = e0;
        afrag[g].y = e1;
    }

    const float* __restrict__ xr   = x   + (size_t)row * TLEN;
    float*       __restrict__ orow = out + (size_t)row * TLEN;

    // uniform recurrence seed: e[-1] = x[0]  (gives e[0] = x[0] as in reference)
    float carry = xr[0];

    for (int tile = 0; tile < TLEN / 256; ++tile) {
        const int base = tile * 256;

        if (tile + 1 < TLEN / 256)
            __builtin_prefetch(xr + base + 256 + lane * 8, 0, 3); // global_prefetch_b8

        // ---- P = M * X via 4 accumulating f32 WMMAs (K = 16) ----
        v8f acc = {0.f, 0.f, 0.f, 0.f, 0.f, 0.f, 0.f, 0.f};
#pragma unroll
        for (int g = 0; g < 4; ++g) {
            const int kk = 4 * g + 2 * half;
            v2f bfrag;
            bfrag.x = xr[base + nn * 16 + kk];
            bfrag.y = xr[base + nn * 16 + kk + 1];
            acc = __builtin_amdgcn_wmma_f32_16x16x4_f32(
                /*neg_a=*/false, afrag[g], /*neg_b=*/false, bfrag,
                /*c_mod=*/(short)0, acc, /*reuse_a=*/false, /*reuse_b=*/false);
        }

        // ---- chunk-carry affine scan: s_n = P[15][n] lives in VGPR7, lane 16+n ----
        float sv = __shfl(acc[7], 16 + nn, 32);   // broadcast s_n to both halves
        float m_acc = d16, a_acc = sv;            // F_n(c) = d16*c + s_n
#pragma unroll
        for (int dlt = 1; dlt < 16; dlt <<= 1) {  // inclusive scan, 16-wide segments
            float mp = __shfl_up(m_acc, dlt, 16);
            float ap = __shfl_up(a_acc, dlt, 16);
            if (nn >= dlt) { a_acc = a_acc + m_acc * ap; m_acc = m_acc * mp; }
        }
        const float t_n   = m_acc * carry + a_acc;          // e at end of chunk n
        const float cprev = __shfl_up(t_n, 1, 16);
        const float c_n   = (nn == 0) ? carry : cprev;      // carry entering chunk n
        const float carry_next = __shfl(t_n, 15, 16);       // e at end of tile

        // ---- elementwise PCEN: lane covers 8 contiguous elems at 16n + 8*half ----
        const float* xp = xr + base + nn * 16 + 8 * half;
        const v4f x0 = *(const v4f*)(xp);
        const v4f x1 = *(const v4f*)(xp + 4);

        v4f o0, o1;
#pragma unroll
        for (int v = 0; v < 8; ++v) {
            const float xe = (v < 4) ? x0[v] : x1[v - 4];
            const float e  = acc[v] + dp[v] * c_n;              // true EMA
            const float lg = flog2(PCEN_EPS + e);
            const float q  = xe * fexp2(-a * lg) + del;         // x/(eps+ema)^a + delta
            const float o  = fexp2(inv_r * flog2(q)) - dpr;     // (...)^(1/r) - delta^(1/r)
            if (v < 4) o0[v] = o; else o1[v - 4] = o;
        }

        float* op = orow + base + nn * 16 + 8 * half;
        *(v4f*)(op)     = o0;
        *(v4f*)(op + 4) = o1;

        carry = carry_next;
    }
}

extern "C" void kernel_launch(void* const* d_in, const int* in_sizes, int n_in,
                              void* d_out, int out_size, void* d_ws, size_t ws_size,
                              hipStream_t stream) {
    const float* x     = (const float*)d_in[0];
    const float* alpha = (const float*)d_in[1];
    const float* delta = (const float*)d_in[2];
    const float* root  = (const float*)d_in[3];
    const float* emaw  = (const float*)d_in[4];
    float* out = (float*)d_out;

    dim3 grid(NROWS / WAVES_PER_BLOCK);   // 512 blocks
    dim3 block(WAVES_PER_BLOCK * 32);     // 256 threads = 8 waves
    hipLaunchKernelGGL(pcen_wmma_kernel, grid, block, 0, stream,
                       x, alpha, delta, root, emaw, out);
}